// BinRegularizer_25572235280639
// MI455X (gfx1250) — compile-verified
//
#include <hip/hip_runtime.h>
#include <hip/hip_bf16.h>

#define NBINS  16
#define CHUNK  1024          // floats per TDM tile (4 KB)
#define WPB    4             // waves per block (wave32 -> 128 threads)
#define GRID   2048          // 8192 wave-streams; 2 tiles each in flight = 64 MB > BDP

typedef unsigned int u32x4 __attribute__((ext_vector_type(4)));
typedef int          i32x8 __attribute__((ext_vector_type(8)));
typedef int          i32x4 __attribute__((ext_vector_type(4)));

// Issue one TDM tile load: nElems f32 from gsrc -> LDS at lds_dst.
// D# per CDNA5 ISA (count=1, type=2, data_size=4B,
// 2D tensor nElems x 1, tile nElems x 1, no pad/iterate/multicast).
__device__ __forceinline__ void tdm_load(const float* gsrc, float* lds_dst, int nElems) {
    unsigned long long ga = (unsigned long long)(uintptr_t)gsrc;
    unsigned int       la = (unsigned int)(uintptr_t)lds_dst;   // flat addr[31:0] == LDS byte offset

    u32x4 g0;
    g0.x = 1u;                                            // count=1, user mode
    g0.y = la;                                            // lds_addr[31:0]
    g0.z = (unsigned int)(ga & 0xFFFFFFFFull);            // global_addr[31:0]
    g0.w = (unsigned int)((ga >> 32) & 0x1FFFFFFull)      // global_addr[56:32]
         | (2u << 30);                                    // type = 2 ("image")

    i32x8 g1;
    g1[0] = (2 << 16);                                    // data_size = 2 -> 4 bytes; wg_mask=0
    g1[1] = (nElems & 0xFFFF) << 16;                      // tensor_dim0[15:0] -> w1[31:16]
    g1[2] = ((nElems >> 16) & 0xFFFF) | (1 << 16);        // tensor_dim0[31:16] | tensor_dim1 lo16 = 1
    g1[3] = (nElems & 0xFFFF) << 16;                      // tensor_dim1 hi16=0 | tile_dim0 = nElems
    g1[4] = 0;                                            // tile_dim1=0 (unused), tile_dim2=0
    g1[5] = nElems;                                       // tensor_dim0_stride[31:0]
    g1[6] = 0;                                            // stride hi / dim1_stride lo
    g1[7] = 0;

    i32x4 z4 = {0, 0, 0, 0};
    i32x8 z8 = {0, 0, 0, 0, 0, 0, 0, 0};
    __builtin_amdgcn_tensor_load_to_lds(g0, g1, z4, z4, z8, 0);  // clang-23 6-arg form
}

__global__ void __launch_bounds__(WPB * 32)
bin_hist_kernel(const float* __restrict__ x, const float* __restrict__ alpha_p,
                float* __restrict__ partial, unsigned int nChunks) {
    __shared__ __align__(16) float stage[WPB][2][CHUNK];          // 32 KB
    __shared__ float acc[WPB * 3 * NBINS * 32];                   // 24 KB, lane-contiguous

    const int tid  = threadIdx.x;
    const int wave = tid >> 5;
    const int lane = tid & 31;

    for (int i = tid; i < WPB * 3 * NBINS * 32; i += blockDim.x) acc[i] = 0.0f;
    __syncthreads();

    const float alpha = *alpha_p;
    const float inv_a = 1.0f / alpha;

    float* accw = &acc[wave * (3 * NBINS * 32)];

    unsigned int c      = (unsigned int)blockIdx.x * WPB + wave;   // nChunks fits in u32
    unsigned int stride = (unsigned int)gridDim.x * WPB;

    int p = 0;
    if (c < nChunks) tdm_load(x + (unsigned long long)c * CHUNK, &stage[wave][0][0], CHUNK);

    while (c < nChunks) {
        unsigned int nxt = c + stride;
        if (nxt < nChunks) {
            tdm_load(x + (unsigned long long)nxt * CHUNK, &stage[wave][p ^ 1][0], CHUNK);
            asm volatile("s_wait_tensorcnt 0x1" ::: "memory");   // oldest tile (ours) done
        } else {
            asm volatile("s_wait_tensorcnt 0x0" ::: "memory");
        }

        const float* buf = &stage[wave][p][0];
#pragma unroll
        for (int j = 0; j < 8; ++j) {
            const float4 v4 = *(const float4*)(buf + lane * 4 + j * 128);
            const float vs[4] = {v4.x, v4.y, v4.z, v4.w};
#pragma unroll
            for (int e = 0; e < 4; ++e) {
                float v = vs[e];
                float t = __builtin_amdgcn_fmed3f(v * inv_a, 0.0f, 15.0f); // v_med3_f32 clamp
                unsigned int bin = (unsigned int)rintf(t);       // v_rndne + v_cvt
                float* slot = accw + bin * 32 + lane;            // 32 distinct banks per wave
                atomicAdd(slot,                     1.0f);       // ds_add_f32 (count)
                atomicAdd(slot + 1 * NBINS * 32,    v);          // ds_add_f32 (sum)
                atomicAdd(slot + 2 * NBINS * 32,    v * v);      // ds_add_f32 (sumsq)
            }
        }
        asm volatile("s_wait_dscnt 0x0" ::: "memory");           // drain reads before TDM reuses buf
        c = nxt;
        p ^= 1;
    }
    __syncthreads();

    // Block reduce: 48 (stat,bin) cells, each summed over WPB waves x 32 lanes.
    for (int idx = tid; idx < 3 * NBINS; idx += blockDim.x) {
        float s = 0.0f;
        for (int w = 0; w < WPB; ++w) {
            const float* a = &acc[w * (3 * NBINS * 32) + idx * 32];
#pragma unroll
            for (int l = 0; l < 32; ++l) s += a[l];
        }
        atomicAdd(&partial[idx], s);
    }
}

__global__ void zero_ws_kernel(float* ws) {
    if (threadIdx.x < 3 * NBINS) ws[threadIdx.x] = 0.0f;
}

__global__ void finalize_kernel(const float* __restrict__ x, const float* __restrict__ alpha_p,
                                const float* __restrict__ ws,
                                unsigned long long tailStart, unsigned long long n,
                                float* __restrict__ out) {
    if (threadIdx.x != 0 || blockIdx.x != 0) return;
    const float alpha = *alpha_p;
    const float inv_a = 1.0f / alpha;

    float cnt[NBINS], sm[NBINS], sq[NBINS];
    for (int i = 0; i < NBINS; ++i) { cnt[i] = ws[i]; sm[i] = ws[NBINS + i]; sq[i] = ws[2 * NBINS + i]; }

    // Generic tail (empty when N % CHUNK == 0, as here: 51,380,224 = 50,176*1024).
    for (unsigned long long i = tailStart; i < n; ++i) {
        float v = x[i];
        float t = __builtin_amdgcn_fmed3f(v * inv_a, 0.0f, 15.0f);
        int b = (int)rintf(t);
        cnt[b] += 1.0f; sm[b] += v; sq[b] += v * v;
    }

    float tmse = 0.0f, tvar = 0.0f;
    for (int i = 0; i < NBINS; ++i) {
        float c    = cnt[i];
        float sc   = fmaxf(c, 1.0f);
        float mean = sm[i] / sc;
        float lvl  = (float)i * alpha;
        if (c > 0.0f)  tmse += (mean - lvl) * (mean - lvl);
        if (c >= 2.0f) tvar += sq[i] / sc - mean * mean;
    }
    out[0] = tmse + tvar;
    out[1] = tmse;
    out[2] = tvar;
}

extern "C" void kernel_launch(void* const* d_in, const int* in_sizes, int n_in,
                              void* d_out, int out_size, void* d_ws, size_t ws_size,
                              hipStream_t stream) {
    const float* x     = (const float*)d_in[0];
    const float* alpha = (const float*)d_in[1];
    float*       out   = (float*)d_out;
    float*       ws    = (float*)d_ws;

    const unsigned long long n       = (unsigned long long)in_sizes[0];
    const unsigned int       nChunks = (unsigned int)(n / CHUNK);

    zero_ws_kernel<<<1, 64, 0, stream>>>(ws);
    if (nChunks > 0)
        bin_hist_kernel<<<GRID, WPB * 32, 0, stream>>>(x, alpha, ws, nChunks);
    finalize_kernel<<<1, 32, 0, stream>>>(x, alpha, ws, (unsigned long long)nChunks * CHUNK, n, out);
}